// FCOSLayer_54623394070751
// MI455X (gfx1250) — compile-verified
//
#include <hip/hip_runtime.h>
#include <math.h>

// FCOS level-0 decode: [32,128,128] locations, 80 classes, stride 8.
// Memory-bound (~190 MB total traffic -> ~8.2 us at 23.3 TB/s HBM).
// CDNA5 path: per-wave async DMA of cls_logits into LDS
// (global_load_async_to_lds_b128 + s_wait_asynccnt), argmax from LDS.

#define NB   32
#define NH   128
#define NW   128
#define NC   80
#define NLOC (NB * NH * NW)          // 524288
#define TPB  128                      // 4 waves per block

#define LOC_BYTES     (NC * 4)        // 320 B of class logits per location
#define LOC_PAD_BYTES 336             // 84 dwords: 16B-aligned, <=2-way LDS bank conflict
#define WAVE_LOCS     32              // one location per lane
#define WAVE_CHUNK    (WAVE_LOCS * LOC_BYTES)   // 10240 B, contiguous in global
#define ASYNC_ITERS   (WAVE_CHUNK / (32 * 16))  // 20 x (32 lanes x 16B = 512B coalesced)

__global__ __launch_bounds__(TPB)
void fcos_decode_kernel(const float* __restrict__ t_ltrb,
                        const float* __restrict__ center_logits,
                        const float* __restrict__ cls_logits,
                        float* __restrict__ out)
{
    __shared__ __align__(16) unsigned char smem[TPB * LOC_PAD_BYTES]; // 43008 B

    const int tid    = (int)threadIdx.x;
    const int lane   = tid & 31;
    const int waveId = tid >> 5;
    const long long g = (long long)blockIdx.x * TPB + tid;   // flat location id

    // ---- Async-DMA this wave's 32 locations of cls_logits into LDS ----
    // Global chunk for this wave is contiguous: 10240 B starting at its first
    // location. Each async b128 instruction moves a contiguous 512 B.
    const unsigned char* gchunk =
        (const unsigned char*)cls_logits +
        ((long long)blockIdx.x * TPB + (long long)waveId * WAVE_LOCS) * (long long)LOC_BYTES;

    const unsigned lds_wave_base =
        (unsigned)(size_t)(&smem[waveId * WAVE_LOCS * LOC_PAD_BYTES]);

#pragma unroll
    for (int i = 0; i < ASYNC_ITERS; ++i) {
        const unsigned off = (unsigned)(i * 512 + lane * 16);   // [0, 10240)
        const unsigned loc = off / LOC_BYTES;                   // which location (0..31)
        const unsigned rem = off - loc * LOC_BYTES;             // byte within location
        const unsigned lds_addr = lds_wave_base + loc * LOC_PAD_BYTES + rem;
        const unsigned char* ga = gchunk + off;
        asm volatile("global_load_async_to_lds_b128 %0, %1, off"
                     :: "v"(lds_addr), "v"(ga)
                     : "memory");
    }

    // ---- Overlap: direct, perfectly-coalesced loads of ltrb + centerness ----
    const float4 lt = ((const float4*)t_ltrb)[g];
    const float  cl = center_logits[g];

    // Wait for this wave's async LDS writes (per-wave staging -> no barrier).
    asm volatile("s_wait_asynccnt 0x0" ::: "memory");

    // ---- argmax over 80 class logits from LDS (first-occurrence tie-break) ----
    const float4* my = (const float4*)(&smem[tid * LOC_PAD_BYTES]);
    float best = -INFINITY;
    int   bidx = 0;
#pragma unroll
    for (int j = 0; j < NC / 4; ++j) {
        const float4 v = my[j];
        if (v.x > best) { best = v.x; bidx = 4 * j + 0; }
        if (v.y > best) { best = v.y; bidx = 4 * j + 1; }
        if (v.z > best) { best = v.z; bidx = 4 * j + 2; }
        if (v.w > best) { best = v.w; bidx = 4 * j + 3; }
    }

    // ---- decode ltrb -> xywh (stride 8, cell centers at +4) ----
    const int wq = (int)(g & (NW - 1));
    const int hq = (int)((g >> 7) & (NH - 1));

    const float l  = __expf(lt.x) * 8.0f;
    const float tp = __expf(lt.y) * 8.0f;
    const float r  = __expf(lt.z) * 8.0f;
    const float bt = __expf(lt.w) * 8.0f;

    const float cx = ((float)wq * 8.0f + 4.0f) + 0.5f * (r - l);
    const float cy = ((float)hq * 8.0f + 4.0f) + 0.5f * (bt - tp);
    const float pw = l + r;
    const float ph = tp + bt;

    // sigmoid is monotonic: max/argmax over sigmoid(logits) == over logits
    const float pc   = 1.0f / (1.0f + __expf(-cl));
    const float ps   = 1.0f / (1.0f + __expf(-best));
    const float conf = sqrtf(pc * ps);

    // ---- outputs: [xywh | cls_idx | conf], concatenated flat ----
    ((float4*)out)[g] = make_float4(cx, cy, pw, ph);
    float* oidx  = out + (long long)NLOC * 4;
    float* oconf = oidx + NLOC;
    oidx[g]  = (float)bidx;
    oconf[g] = conf;
}

extern "C" void kernel_launch(void* const* d_in, const int* in_sizes, int n_in,
                              void* d_out, int out_size, void* d_ws, size_t ws_size,
                              hipStream_t stream)
{
    (void)in_sizes; (void)n_in; (void)d_ws; (void)ws_size; (void)out_size;

    const float* t_ltrb = (const float*)d_in[0];
    const float* center = (const float*)d_in[1];
    const float* cls    = (const float*)d_in[2];
    // d_in[3] = img_h, d_in[4] = img_w : unused by the reference math
    float* out = (float*)d_out;

    const dim3 grid(NLOC / TPB);   // 4096 blocks, exact cover
    const dim3 block(TPB);
    hipLaunchKernelGGL(fcos_decode_kernel, grid, block, 0, stream,
                       t_ltrb, center, cls, out);
}